// SimplifiedRGNN_3865470567041
// MI455X (gfx1250) — compile-verified
//
#include <hip/hip_runtime.h>
#include <hip/hip_bf16.h>
#include <math.h>

typedef __attribute__((ext_vector_type(2))) float v2f;
typedef __attribute__((ext_vector_type(8))) float v8f;

static __host__ int cdiv(long long a, long long b) { return (int)((a + b - 1) / b); }

// ---------------------------------------------------------------------------
// WMMA f32 GEMM: out[n x 64] = act( A[n x K] @ W[K x 64] + bias )
// A columns 0..63 come from A1, columns 64..127 (if K==128) from A2.
// Each wave computes a 16x16 tile via V_WMMA_F32_16X16X4_F32; block = 8 waves
// covering a 32x64 tile. W (and bias) staged in LDS.
// ---------------------------------------------------------------------------
__global__ __launch_bounds__(256)
void wmma_gemm64(const float* __restrict__ A1, const float* __restrict__ A2,
                 const float* __restrict__ W, const float* __restrict__ bias,
                 float* __restrict__ out, int n, int K, float slope, int applyAct)
{
    __shared__ float sW[128 * 64];
    __shared__ float sB[64];

    for (int i = threadIdx.x; i < K * 16; i += 256)
        ((float4*)sW)[i] = ((const float4*)W)[i];
    if (threadIdx.x < 64)
        sB[threadIdx.x] = bias ? bias[threadIdx.x] : 0.0f;
    __syncthreads();

    const int wave = threadIdx.x >> 5;
    const int lane = threadIdx.x & 31;
    const int hf   = lane >> 4;        // which 16-lane half
    const int lm   = lane & 15;
    const int r0   = blockIdx.x * 32 + (wave >> 2) * 16;
    const int c0   = (wave & 3) * 16;

    int ar = r0 + lm; if (ar > n - 1) ar = n - 1;   // clamp (keeps EXEC full)

    v8f acc = {};
    for (int k = 0; k < K; k += 4) {
        const int kk = k + hf * 2;     // this half's two consecutive K values
        // A fragment: lane = M row, (K=kk, kk+1)
        const float* ap = (kk < 64) ? (A1 + (size_t)ar * 64 + kk)
                                    : (A2 + (size_t)ar * 64 + (kk - 64));
        v2f a = *(const v2f*)ap;
        // B fragment: lane = N col, (K=kk, kk+1)
        v2f b;
        b.x = sW[kk * 64 + c0 + lm];
        b.y = sW[(kk + 1) * 64 + c0 + lm];
        acc = __builtin_amdgcn_wmma_f32_16x16x4_f32(
            /*neg_a=*/false, a, /*neg_b=*/false, b,
            /*c_mod=*/(short)0, acc, /*reuse_a=*/false, /*reuse_b=*/false);
    }

    const int col = c0 + lm;
#pragma unroll
    for (int v = 0; v < 8; ++v) {
        int row = r0 + (hf << 3) + v;  // M = v (+8 for upper half)
        if (row < n) {
            float val = acc[v] + sB[col];
            if (applyAct) val = (val >= 0.0f) ? val : slope * val;
            out[(size_t)row * 64 + col] = val;
        }
    }
}

// ---------------------------------------------------------------------------
// a_s[i] = dot(hw[i,:], att_src), a_d[i] = dot(hw[i,:], att_dst)
// One wave32 per node: 2 floats/lane, shfl-xor tree reduction.
// ---------------------------------------------------------------------------
__global__ __launch_bounds__(256)
void att_kernel(const float* __restrict__ hw, const float* __restrict__ atts,
                const float* __restrict__ attd, float* __restrict__ aS,
                float* __restrict__ aD, int n)
{
    int gid  = blockIdx.x * blockDim.x + threadIdx.x;
    int node = gid >> 5;
    int lane = gid & 31;
    if (node >= n) return;
    float2 v  = *(const float2*)(hw + (size_t)node * 64 + lane * 2);
    float2 s2 = *(const float2*)(atts + lane * 2);
    float2 d2 = *(const float2*)(attd + lane * 2);
    float s = v.x * s2.x + v.y * s2.y;
    float d = v.x * d2.x + v.y * d2.y;
#pragma unroll
    for (int off = 16; off >= 1; off >>= 1) {
        s += __shfl_xor(s, off, 32);
        d += __shfl_xor(d, off, 32);
    }
    if (lane == 0) { aS[node] = s; aD[node] = d; }
}

__global__ __launch_bounds__(256)
void init_node(unsigned* __restrict__ menc, float* __restrict__ den, int n)
{
    int i = blockIdx.x * blockDim.x + threadIdx.x;
    if (i < n) { menc[i] = 0u; den[i] = 0.0f; }
}

// Seed the scatter target with the bias so no separate bias pass is needed.
__global__ __launch_bounds__(256)
void init_out(float* __restrict__ out, const float* __restrict__ bias, int n)
{
    int i = blockIdx.x * blockDim.x + threadIdx.x;
    if (i < n * 64) out[i] = bias[i & 63];
}

__device__ __forceinline__ float edge_logit(int e, int E, const int* __restrict__ src,
                                            const int* __restrict__ dst,
                                            const float* __restrict__ aS,
                                            const float* __restrict__ aD,
                                            int* s, int* d)
{
    int ss, dd;
    if (e < E) { ss = src[e]; dd = dst[e]; } else { ss = dd = e - E; }  // self loops
    *s = ss; *d = dd;
    float l = aS[ss] + aD[dd];
    return (l >= 0.0f) ? l : 0.2f * l;   // leaky_relu, slope 0.2
}

// Order-preserving float->uint encoding so segment-max can use atomicMax(u32).
__global__ __launch_bounds__(256)
void edge_max(const int* __restrict__ src, const int* __restrict__ dst,
              const float* __restrict__ aS, const float* __restrict__ aD,
              unsigned* __restrict__ menc, int E, int EN)
{
    int e = blockIdx.x * blockDim.x + threadIdx.x;
    if (e >= EN) return;
    int s, d;
    float l = edge_logit(e, E, src, dst, aS, aD, &s, &d);
    unsigned bits = __float_as_uint(l);
    unsigned enc  = (bits & 0x80000000u) ? ~bits : (bits | 0x80000000u);
    atomicMax(menc + d, enc);
}

__global__ __launch_bounds__(256)
void m_final(unsigned* __restrict__ menc, float* __restrict__ m, int n)
{
    int i = blockIdx.x * blockDim.x + threadIdx.x;
    if (i >= n) return;
    unsigned u    = menc[i];
    unsigned bits = (u & 0x80000000u) ? (u & 0x7FFFFFFFu) : ~u;
    float f = __uint_as_float(bits);
    if (!isfinite(f)) f = 0.0f;     // matches reference's where(isfinite, m, 0)
    m[i] = f;
}

__global__ __launch_bounds__(256)
void edge_exp(const int* __restrict__ src, const int* __restrict__ dst,
              const float* __restrict__ aS, const float* __restrict__ aD,
              const float* __restrict__ m, float* __restrict__ ex,
              float* __restrict__ den, int E, int EN)
{
    int e = blockIdx.x * blockDim.x + threadIdx.x;
    if (e >= EN) return;
    int s, d;
    float l = edge_logit(e, E, src, dst, aS, aD, &s, &d);
    float v = expf(l - m[d]);
    ex[e] = v;
    atomicAdd(den + d, v);
}

// Heavy pass: each 16-lane group handles one edge x 64 channels.
// float4 (global_load_b128) gathers + 4x global_atomic_add_f32 scatters.
__global__ __launch_bounds__(256)
void scatter_kernel(const int* __restrict__ src, const int* __restrict__ dst,
                    const float* __restrict__ ex, const float* __restrict__ den,
                    const float* __restrict__ hw, float* __restrict__ out,
                    int E, int EN)
{
    long long t = (long long)blockIdx.x * blockDim.x + threadIdx.x;
    if (t >= (long long)EN * 16) return;
    int e  = (int)(t >> 4);
    int cg = ((int)t & 15) << 2;
    int s, d;
    if (e < E) { s = src[e]; d = dst[e]; } else { s = d = e - E; }
    float alpha = ex[e] / (den[d] + 1e-16f);
    float4 v = *(const float4*)(hw + (size_t)s * 64 + cg);
    float* o = out + (size_t)d * 64 + cg;
    atomicAdd(o + 0, alpha * v.x);
    atomicAdd(o + 1, alpha * v.y);
    atomicAdd(o + 2, alpha * v.z);
    atomicAdd(o + 3, alpha * v.w);
}

// ---------------------------------------------------------------------------
extern "C" void kernel_launch(void* const* d_in, const int* in_sizes, int n_in,
                              void* d_out, int out_size, void* d_ws, size_t ws_size,
                              hipStream_t stream)
{
    const float* x    = (const float*)d_in[0];
    const int*   ei   = (const int*)  d_in[1];
    const float* rgx  = (const float*)d_in[3];
    const int*   rei  = (const int*)  d_in[4];
    const float* W1   = (const float*)d_in[6];
    const float* b1   = (const float*)d_in[7];
    const float* aW   = (const float*)d_in[8];
    const float* aAS  = (const float*)d_in[9];
    const float* aAD  = (const float*)d_in[10];
    const float* aB   = (const float*)d_in[11];
    const float* rW   = (const float*)d_in[12];
    const float* rAS  = (const float*)d_in[13];
    const float* rAD  = (const float*)d_in[14];
    const float* rB   = (const float*)d_in[15];
    const float* mW   = (const float*)d_in[16];
    const float* mAS  = (const float*)d_in[17];
    const float* mAD  = (const float*)d_in[18];
    const float* mB   = (const float*)d_in[19];
    const float* W2   = (const float*)d_in[20];
    const float* b2   = (const float*)d_in[21];

    const int N  = in_sizes[0] / 64;   // 100000
    const int E  = in_sizes[1] / 2;    // 1200000
    const int EN = E + N;

    // Workspace layout (floats)
    const size_t NC = (size_t)N * 64;
    float* B0  = (float*)d_ws;         // lin1 outputs (reused)
    float* B1  = B0 + NC;              // atom conv out
    float* B2  = B1 + NC;              // rg conv out
    float* B3  = B2 + NC;              // mol conv out
    float* HW  = B3 + NC;              // per-conv h@W scratch
    float* aS  = HW + NC;
    float* aD  = aS + N;
    float* Mb  = aD + N;               // max buffer (u32 enc, then f32)
    float* Den = Mb + N;
    float* Ex  = Den + N;              // [E+N]

    const int gGemm = cdiv(N, 32);

    auto gat = [&](const float* xin, const int* srcp, const int* dstp,
                   const float* Wc, const float* atts, const float* attd,
                   const float* biasc, float* outp) {
        wmma_gemm64<<<gGemm, 256, 0, stream>>>(xin, nullptr, Wc, nullptr, HW, N, 64, 0.0f, 0);
        att_kernel<<<cdiv((long long)N * 32, 256), 256, 0, stream>>>(HW, atts, attd, aS, aD, N);
        init_node<<<cdiv(N, 256), 256, 0, stream>>>((unsigned*)Mb, Den, N);
        init_out<<<cdiv((long long)N * 64, 256), 256, 0, stream>>>(outp, biasc, N);
        edge_max<<<cdiv(EN, 256), 256, 0, stream>>>(srcp, dstp, aS, aD, (unsigned*)Mb, E, EN);
        m_final<<<cdiv(N, 256), 256, 0, stream>>>((unsigned*)Mb, Mb, N);
        edge_exp<<<cdiv(EN, 256), 256, 0, stream>>>(srcp, dstp, aS, aD, Mb, Ex, Den, E, EN);
        scatter_kernel<<<cdiv((long long)EN * 16, 256), 256, 0, stream>>>(
            srcp, dstp, Ex, Den, HW, outp, E, EN);
    };

    // h = leaky(x @ W1 + b1, 0.01)
    wmma_gemm64<<<gGemm, 256, 0, stream>>>(x, nullptr, W1, b1, B0, N, 64, 0.01f, 1);
    // h1 = atom GAT(h)
    gat(B0, ei, ei + E, aW, aAS, aAD, aB, B1);
    // rh = leaky(rg_x @ W1 + b1, 0.01)   (B0 reusable now)
    wmma_gemm64<<<gGemm, 256, 0, stream>>>(rgx, nullptr, W1, b1, B0, N, 64, 0.01f, 1);
    // rh1 = rg GAT(rh)
    gat(B0, rei, rei + E, rW, rAS, rAD, rB, B2);
    // out_a = mol GAT(h1)
    gat(B1, ei, ei + E, mW, mAS, mAD, mB, B3);
    // d_out = concat(out_a, rh1) @ W2 + b2  (split-K, K=128)
    wmma_gemm64<<<gGemm, 256, 0, stream>>>(B3, B2, W2, b2, (float*)d_out, N, 128, 0.0f, 0);
}